// GNNCell_22952305230188
// MI455X (gfx1250) — compile-verified
//
#include <hip/hip_runtime.h>
#include <hip/hip_bf16.h>

typedef float v2f __attribute__((ext_vector_type(2)));
typedef float v4f __attribute__((ext_vector_type(4)));
typedef float v8f __attribute__((ext_vector_type(8)));

#define GB 512   // batch
#define GN 512   // nodes / input dim / hidden state size
#define GH 256   // hidden_dim

// ---------------------------------------------------------------------------
// Generic transpose: out[c*R + r] = in[r*C + c]   (in: [R,C] row-major)
// Weights are tiny (<=3 MB) and L2-resident; simple per-element is fine.
// ---------------------------------------------------------------------------
__global__ void transpose_kernel(const float* __restrict__ in, float* __restrict__ out,
                                 int R, int C) {
    int idx = blockIdx.x * blockDim.x + threadIdx.x;
    if (idx >= R * C) return;
    int r = idx / C;
    int c = idx - r * C;
    out[(size_t)c * R + r] = in[idx];
}

// ---------------------------------------------------------------------------
// agg[b,n] = sum_m adj[b,n,m] * x[b,m]
// One wave32 per row of adj. 536 MB streamed once -> non-temporal b128 loads.
// x[b] (2 KB) is reused by 512 rows and stays in cache.
// ---------------------------------------------------------------------------
__global__ void agg_kernel(const float* __restrict__ adj, const float* __restrict__ x,
                           float* __restrict__ agg) {
    const int wave = blockIdx.x * (blockDim.x >> 5) + (threadIdx.x >> 5);
    const int lane = threadIdx.x & 31;
    const int b = wave >> 9;          // / GN
    const int n = wave & (GN - 1);

    const v4f* __restrict__ row = (const v4f*)(adj + ((size_t)b * GN + n) * GN);
    const v4f* __restrict__ xv  = (const v4f*)(x + (size_t)b * GN);

    float acc = 0.0f;
#pragma unroll
    for (int i = 0; i < GN / (32 * 4); ++i) {
        v4f a  = __builtin_nontemporal_load(&row[i * 32 + lane]);
        v4f xx = xv[i * 32 + lane];
        acc += a.x * xx.x + a.y * xx.y + a.z * xx.z + a.w * xx.w;
    }
#pragma unroll
    for (int m = 16; m >= 1; m >>= 1)
        acc += __shfl_xor(acc, m, 32);

    if (lane == 0) agg[wave] = acc;
}

// ---------------------------------------------------------------------------
// f32 WMMA GEMM: C[M,N] = act(A[M,K] * Bt[K,N] + bias[N])
//   A  : row-major [M,K]
//   Bt : row-major [K,N]  (pre-transposed weights -> lane-coalesced loads)
// One wave per 16x16 C tile, K consumed 4 at a time via V_WMMA_F32_16X16X4_F32.
// f32 A-fragment layout (ISA 7.12.2): lanes 0-15 hold K={0,1}, lanes 16-31 K={2,3}.
// C/D: VGPR i holds row M = i + 8*half, col = lane&15.
// ---------------------------------------------------------------------------
__global__ void wmma_gemm_f32(const float* __restrict__ A, const float* __restrict__ Bt,
                              const float* __restrict__ bias, float* __restrict__ C,
                              int M, int N, int K, int act) {
    const int wave = blockIdx.x * (blockDim.x >> 5) + (threadIdx.x >> 5);
    const int lane = threadIdx.x & 31;
    const int tilesN = N >> 4;
    const int tm = (wave / tilesN) << 4;
    const int tn = (wave % tilesN) << 4;
    if (tm >= M) return;                       // wave-uniform: EXEC stays all-1

    const int l    = lane & 15;
    const int half = lane >> 4;

    const float* __restrict__ Arow = A  + (size_t)(tm + l) * K + 2 * half;
    const float* __restrict__ Bcol = Bt + (size_t)(2 * half) * N + tn + l;

    v8f acc = {};
    for (int k = 0; k < K; k += 4) {
        v2f a = *(const v2f*)(Arow + k);       // A[tm+l, k+2*half .. +1]
        v2f b;
        b.x = Bcol[(size_t)k * N];             // Bt[k+2*half,     tn+l]
        b.y = Bcol[(size_t)(k + 1) * N];       // Bt[k+2*half + 1, tn+l]
        acc = __builtin_amdgcn_wmma_f32_16x16x4_f32(
            /*neg_a=*/false, a, /*neg_b=*/false, b,
            /*c_mod=*/(short)0, acc, /*reuse_a=*/false, /*reuse_b=*/false);
    }

    const float bv = bias ? bias[tn + l] : 0.0f;
#pragma unroll
    for (int i = 0; i < 8; ++i) {
        int row = tm + i + 8 * half;
        float v = acc[i] + bv;
        if (act) v = fmaxf(v, 0.0f);
        C[(size_t)row * N + tn + l] = v;
    }
}

// ---------------------------------------------------------------------------
// GRU gate fusion (PyTorch order r,z,n over column blocks of 3N):
//   r = sig(gi_r+gh_r); z = sig(gi_z+gh_z); nn = tanh(gi_n + r*gh_n)
//   out = (1-z)*nn + z*x
// ---------------------------------------------------------------------------
__global__ void gates_kernel(const float* __restrict__ gi, const float* __restrict__ gh,
                             const float* __restrict__ x, float* __restrict__ out) {
    int idx = blockIdx.x * blockDim.x + threadIdx.x;
    if (idx >= GB * GN) return;
    int b = idx >> 9;
    int n = idx & (GN - 1);
    const float* gib = gi + (size_t)b * 3 * GN;
    const float* ghb = gh + (size_t)b * 3 * GN;

    float r  = 1.0f / (1.0f + expf(-(gib[n]          + ghb[n])));
    float z  = 1.0f / (1.0f + expf(-(gib[GN + n]     + ghb[GN + n])));
    float nn = tanhf(gib[2 * GN + n] + r * ghb[2 * GN + n]);
    out[idx] = (1.0f - z) * nn + z * x[idx];
}

extern "C" void kernel_launch(void* const* d_in, const int* in_sizes, int n_in,
                              void* d_out, int out_size, void* d_ws, size_t ws_size,
                              hipStream_t stream) {
    const float* x     = (const float*)d_in[0];   // [B,N]
    const float* adj   = (const float*)d_in[1];   // [B,N,N]
    const float* W_msg = (const float*)d_in[2];   // [H,N]
    const float* b_msg = (const float*)d_in[3];   // [H]
    const float* W_ih  = (const float*)d_in[4];   // [3N,H]
    const float* b_ih  = (const float*)d_in[5];   // [3N]
    const float* W_hh  = (const float*)d_in[6];   // [3N,N]
    const float* b_hh  = (const float*)d_in[7];   // [3N]
    float* out = (float*)d_out;                   // [B,N]

    // Workspace layout (floats)
    float* ws   = (float*)d_ws;
    float* agg  = ws;                                   // B*N      = 262144
    float* m    = agg + (size_t)GB * GN;                // B*H      = 131072
    float* gi   = m   + (size_t)GB * GH;                // B*3N     = 786432
    float* gh   = gi  + (size_t)GB * 3 * GN;            // B*3N     = 786432
    float* WmT  = gh  + (size_t)GB * 3 * GN;            // N*H      = 131072
    float* WiT  = WmT + (size_t)GN * GH;                // H*3N     = 393216
    float* WhT  = WiT + (size_t)GH * 3 * GN;            // N*3N     = 786432

    const int TB = 256;

    // 1) Transpose weights into [K,N] row-major for coalesced WMMA B-loads
    transpose_kernel<<<(GH * GN + TB - 1) / TB, TB, 0, stream>>>(W_msg, WmT, GH, GN);
    transpose_kernel<<<(3 * GN * GH + TB - 1) / TB, TB, 0, stream>>>(W_ih, WiT, 3 * GN, GH);
    transpose_kernel<<<(3 * GN * GN + TB - 1) / TB, TB, 0, stream>>>(W_hh, WhT, 3 * GN, GN);

    // 2) Neighbor aggregation: bandwidth-bound stream of adj (one wave per row)
    {
        int waves  = GB * GN;              // 262144 rows
        int blocks = waves / (TB / 32);    // 8 waves per block
        agg_kernel<<<blocks, TB, 0, stream>>>(adj, x, agg);
    }

    // 3) m = relu(agg @ W_msg^T + b_msg)   [512,512]x[512,256]
    {
        int tiles  = (GB / 16) * (GH / 16);
        wmma_gemm_f32<<<(tiles + 7) / 8, TB, 0, stream>>>(agg, WmT, b_msg, m,
                                                          GB, GH, GN, /*relu*/1);
    }
    // 4) gi = m @ W_ih^T + b_ih            [512,256]x[256,1536]
    {
        int tiles  = (GB / 16) * (3 * GN / 16);
        wmma_gemm_f32<<<(tiles + 7) / 8, TB, 0, stream>>>(m, WiT, b_ih, gi,
                                                          GB, 3 * GN, GH, 0);
    }
    // 5) gh = x @ W_hh^T + b_hh            [512,512]x[512,1536]
    {
        int tiles  = (GB / 16) * (3 * GN / 16);
        wmma_gemm_f32<<<(tiles + 7) / 8, TB, 0, stream>>>(x, WhT, b_hh, gh,
                                                          GB, 3 * GN, GN, 0);
    }

    // 6) GRU gates + blend
    gates_kernel<<<(GB * GN + TB - 1) / TB, TB, 0, stream>>>(gi, gh, x, out);
}